// RNN_GRU_LSTM_49821620633815
// MI455X (gfx1250) — compile-verified
//
#include <hip/hip_runtime.h>
#include <math.h>

// BiLSTM, T=131072, H=32, IN=OUT=1, batch=1, all f32.
// Critical path = serial scans; GEMM parts use v_wmma_f32_16x16x4_f32.

typedef __attribute__((ext_vector_type(2))) float v2f;
typedef __attribute__((ext_vector_type(8))) float v8f;

__device__ __forceinline__ float sigm(float x) {
  return 1.0f / (1.0f + __expf(-x));
}
__device__ __forceinline__ float tanh_fast(float x) {
  x = fminf(12.0f, fmaxf(-12.0f, x));       // avoid inf/inf -> NaN
  float e = __expf(2.0f * x);
  return (e - 1.0f) / (e + 1.0f);
}

// ---------------------------------------------------------------------------
// Layer 0 scan. grid=(2), block=(32). block 0 = forward, block 1 = backward.
// One wave owns the whole recurrence: lane j owns gate rows {j, j+32, j+64,
// j+96} of Whh in registers; h broadcast via __shfl (no LDS, no barriers).
// Writes y0[t*64 + dir*32 + j] and final h,c into d_out tail.
// ---------------------------------------------------------------------------
__global__ void __launch_bounds__(32) lstm_l0_scan(
    const float* __restrict__ x,
    const float* __restrict__ h_state, const float* __restrict__ c_state,
    const float* __restrict__ Wih0f, const float* __restrict__ Whh0f,
    const float* __restrict__ bih0f, const float* __restrict__ bhh0f,
    const float* __restrict__ Wih0b, const float* __restrict__ Whh0b,
    const float* __restrict__ bih0b, const float* __restrict__ bhh0b,
    float* __restrict__ y0, float* __restrict__ outTail, int T)
{
  const int dir = blockIdx.x;
  const int j   = threadIdx.x;
  const float* Wih = dir ? Wih0b : Wih0f;
  const float* Whh = dir ? Whh0b : Whh0f;
  const float* bih = dir ? bih0b : bih0f;
  const float* bhh = dir ? bhh0b : bhh0f;

  float wi[32], wf[32], wg[32], wo[32];
  #pragma unroll
  for (int k = 0; k < 32; ++k) {
    wi[k] = Whh[(j      ) * 32 + k];
    wf[k] = Whh[(j +  32) * 32 + k];
    wg[k] = Whh[(j +  64) * 32 + k];
    wo[k] = Whh[(j +  96) * 32 + k];
  }
  const float ui = Wih[j], uf = Wih[j + 32], ug = Wih[j + 64], uo = Wih[j + 96];
  const float bi = bih[j]      + bhh[j];
  const float bf = bih[j + 32] + bhh[j + 32];
  const float bg = bih[j + 64] + bhh[j + 64];
  const float bo = bih[j + 96] + bhh[j + 96];

  float h = h_state[dir * 32 + j];
  float c = c_state[dir * 32 + j];

  for (int s = 0; s < T; ++s) {
    const int t = dir ? (T - 1 - s) : s;
    if (s + 64 < T) {
      const int tp = dir ? (t - 64) : (t + 64);
      __builtin_prefetch(x + tp, 0, 1);     // global_prefetch_b8
    }
    const float xt = x[t];
    float gi = fmaf(xt, ui, bi);
    float gf = fmaf(xt, uf, bf);
    float gg = fmaf(xt, ug, bg);
    float go = fmaf(xt, uo, bo);
    #pragma unroll
    for (int k = 0; k < 32; ++k) {
      const float hk = __shfl(h, k, 32);
      gi = fmaf(wi[k], hk, gi);
      gf = fmaf(wf[k], hk, gf);
      gg = fmaf(wg[k], hk, gg);
      go = fmaf(wo[k], hk, go);
    }
    const float iv = sigm(gi), fv = sigm(gf);
    const float gv = tanh_fast(gg), ov = sigm(go);
    c = fmaf(fv, c, iv * gv);
    h = ov * tanh_fast(c);
    y0[(size_t)t * 64 + dir * 32 + j] = h;
  }
  outTail[dir * 32 + j]       = h;   // hN slots 0,1
  outTail[128 + dir * 32 + j] = c;   // cN slots 0,1
}

// ---------------------------------------------------------------------------
// Layer-1 input projection (both directions):
//   xg1[t, n] = sum_k y0[t,k] * Wih1[n,k] + bih1[n] + bhh1[n]
// One wave per 16(M=time) x 16(N=gate) tile, K=64 in 16 WMMA f32 16x16x4 steps.
// grid = 2 * (T/16) * 8 = T blocks of 32 threads.
// A layout (16x4 f32): lane L, m=L%16: v0=A[m][2*(L/16)], v1=A[m][2*(L/16)+1]
// B layout (4x16 f32): lane L, n=L%16: v0=B[2*(L/16)][n], v1=B[2*(L/16)+1][n]
// D layout (16x16 f32): VGPR r holds row m = r + 8*(L/16), col n = L%16
// ---------------------------------------------------------------------------
__global__ void __launch_bounds__(32) gemm_xg1(
    const float* __restrict__ y0,
    const float* __restrict__ Wih1f, const float* __restrict__ bih1f,
    const float* __restrict__ bhh1f,
    const float* __restrict__ Wih1b, const float* __restrict__ bih1b,
    const float* __restrict__ bhh1b,
    float* __restrict__ xg1f, float* __restrict__ xg1b, int T)
{
  unsigned tile = blockIdx.x;
  const int dir = tile & 1;  tile >>= 1;
  const int ntile = tile & 7;                 // 128/16 N tiles
  const int mtile = tile >> 3;                // T/16 M tiles
  const float* W   = dir ? Wih1b : Wih1f;
  const float* bi  = dir ? bih1b : bih1f;
  const float* bh  = dir ? bhh1b : bhh1f;
  float*       dst = dir ? xg1b  : xg1f;

  const int lane = threadIdx.x;
  const int half = lane >> 4;                 // 0 | 1
  const int l16  = lane & 15;
  const int m0 = mtile * 16;
  const int n0 = ntile * 16;

  v8f acc = {};
  #pragma unroll
  for (int k0 = 0; k0 < 64; k0 += 4) {
    // A tile: y0 rows m0..m0+15, cols k0..k0+3  (row-major, 8B aligned pairs)
    const float* arow = y0 + (size_t)(m0 + l16) * 64 + k0 + 2 * half;
    v2f a = *reinterpret_cast<const v2f*>(arow);
    // B tile: B[k][n] = Wih1[n0+n][k]  (column of W, W row-major 128x64)
    const float* bcol = W + (size_t)(n0 + l16) * 64 + k0 + 2 * half;
    v2f b = *reinterpret_cast<const v2f*>(bcol);
    acc = __builtin_amdgcn_wmma_f32_16x16x4_f32(
        /*neg_a=*/false, a, /*neg_b=*/false, b,
        /*c_mod=*/(short)0, acc, /*reuse_a=*/false, /*reuse_b=*/false);
  }
  const float bias = bi[n0 + l16] + bh[n0 + l16];   // fold both biases
  #pragma unroll
  for (int r = 0; r < 8; ++r) acc[r] += bias;
  #pragma unroll
  for (int r = 0; r < 8; ++r)
    dst[(size_t)(m0 + r + 8 * half) * 128 + n0 + l16] = acc[r];
}

// ---------------------------------------------------------------------------
// Layer 1 scan. grid=(2), block=(32). Gates come precomputed from xg1
// (bias-folded); only the recurrent 128x32 GEMV per step remains.
// Stores h per step into h1{f,b} for the off-critical-path output projection.
// ---------------------------------------------------------------------------
__global__ void __launch_bounds__(32) lstm_l1_scan(
    const float* __restrict__ xg1f, const float* __restrict__ xg1b,
    const float* __restrict__ h_state, const float* __restrict__ c_state,
    const float* __restrict__ Whh1f, const float* __restrict__ Whh1b,
    float* __restrict__ h1f, float* __restrict__ h1b,
    float* __restrict__ outTail, int T)
{
  const int dir = blockIdx.x;
  const int j   = threadIdx.x;
  const float* xg  = dir ? xg1b : xg1f;
  const float* Whh = dir ? Whh1b : Whh1f;
  float*       hbf = dir ? h1b : h1f;

  float wi[32], wf[32], wg[32], wo[32];
  #pragma unroll
  for (int k = 0; k < 32; ++k) {
    wi[k] = Whh[(j      ) * 32 + k];
    wf[k] = Whh[(j +  32) * 32 + k];
    wg[k] = Whh[(j +  64) * 32 + k];
    wo[k] = Whh[(j +  96) * 32 + k];
  }

  float h = h_state[(2 + dir) * 32 + j];
  float c = c_state[(2 + dir) * 32 + j];

  for (int s = 0; s < T; ++s) {
    const int t = dir ? (T - 1 - s) : s;
    if (s + 8 < T) {
      const int tp = dir ? (t - 8) : (t + 8);
      __builtin_prefetch(xg + (size_t)tp * 128 + j, 0, 1);  // global_prefetch_b8
    }
    const float* g = xg + (size_t)t * 128;
    float gi = g[j];
    float gf = g[j + 32];
    float gg = g[j + 64];
    float go = g[j + 96];
    #pragma unroll
    for (int k = 0; k < 32; ++k) {
      const float hk = __shfl(h, k, 32);
      gi = fmaf(wi[k], hk, gi);
      gf = fmaf(wf[k], hk, gf);
      gg = fmaf(wg[k], hk, gg);
      go = fmaf(wo[k], hk, go);
    }
    const float iv = sigm(gi), fv = sigm(gf);
    const float gv = tanh_fast(gg), ov = sigm(go);
    c = fmaf(fv, c, iv * gv);
    h = ov * tanh_fast(c);
    hbf[(size_t)t * 32 + j] = h;
  }
  outTail[64 + dir * 32 + j]        = h;   // hN slots 2,3
  outTail[128 + 64 + dir * 32 + j]  = c;   // cN slots 2,3
}

// ---------------------------------------------------------------------------
// Output projection: out[t] = bout + Wout[0:32].h1f[t] + Wout[32:64].h1b[t]
// ---------------------------------------------------------------------------
__global__ void __launch_bounds__(256) out_proj(
    const float* __restrict__ h1f, const float* __restrict__ h1b,
    const float* __restrict__ Wout, const float* __restrict__ bout,
    float* __restrict__ out, int T)
{
  const int t = blockIdx.x * 256 + threadIdx.x;
  if (t >= T) return;
  float acc = bout[0];
  const float* a = h1f + (size_t)t * 32;
  const float* b = h1b + (size_t)t * 32;
  #pragma unroll
  for (int k = 0; k < 32; ++k) acc = fmaf(Wout[k], a[k], acc);
  #pragma unroll
  for (int k = 0; k < 32; ++k) acc = fmaf(Wout[32 + k], b[k], acc);
  out[t] = acc;
}

extern "C" void kernel_launch(void* const* d_in, const int* in_sizes, int n_in,
                              void* d_out, int out_size, void* d_ws, size_t ws_size,
                              hipStream_t stream) {
  const float* x       = (const float*)d_in[0];
  const float* h_state = (const float*)d_in[1];
  const float* c_state = (const float*)d_in[2];
  const float* Wih0f   = (const float*)d_in[3];
  const float* Whh0f   = (const float*)d_in[4];
  const float* bih0f   = (const float*)d_in[5];
  const float* bhh0f   = (const float*)d_in[6];
  const float* Wih0b   = (const float*)d_in[7];
  const float* Whh0b   = (const float*)d_in[8];
  const float* bih0b   = (const float*)d_in[9];
  const float* bhh0b   = (const float*)d_in[10];
  const float* Wih1f   = (const float*)d_in[11];
  const float* Whh1f   = (const float*)d_in[12];
  const float* bih1f   = (const float*)d_in[13];
  const float* bhh1f   = (const float*)d_in[14];
  const float* Wih1b   = (const float*)d_in[15];
  const float* Whh1b   = (const float*)d_in[16];
  const float* bih1b   = (const float*)d_in[17];
  const float* bhh1b   = (const float*)d_in[18];
  const float* Wout    = (const float*)d_in[19];
  const float* bout    = (const float*)d_in[20];
  float* out = (float*)d_out;
  const int T = in_sizes[0];   // 131072 (multiple of 16)

  // Workspace layout (floats): y0 | xg1f | xg1b | h1f | h1b
  float* ws   = (float*)d_ws;
  float* y0   = ws;                          // T*64
  float* xg1f = y0   + (size_t)T * 64;       // T*128
  float* xg1b = xg1f + (size_t)T * 128;      // T*128
  float* h1f  = xg1b + (size_t)T * 128;      // T*32
  float* h1b  = h1f  + (size_t)T * 32;       // T*32

  // 1) Layer-0 scans (fwd & bwd run concurrently as two workgroups)
  lstm_l0_scan<<<dim3(2), dim3(32), 0, stream>>>(
      x, h_state, c_state,
      Wih0f, Whh0f, bih0f, bhh0f,
      Wih0b, Whh0b, bih0b, bhh0b,
      y0, out + T, T);

  // 2) Layer-1 input projection GEMM via WMMA: 2 dirs * (T/16) * 8 = T tiles
  gemm_xg1<<<dim3((unsigned)T), dim3(32), 0, stream>>>(
      y0, Wih1f, bih1f, bhh1f, Wih1b, bih1b, bhh1b, xg1f, xg1b, T);

  // 3) Layer-1 scans
  lstm_l1_scan<<<dim3(2), dim3(32), 0, stream>>>(
      xg1f, xg1b, h_state, c_state, Whh1f, Whh1b, h1f, h1b, out + T, T);

  // 4) Output projection
  out_proj<<<dim3((T + 255) / 256), dim3(256), 0, stream>>>(
      h1f, h1b, Wout, bout, out, T);
}